// SimplicialEGNNLayer_14886356648021
// MI455X (gfx1250) — compile-verified
//
#include <hip/hip_runtime.h>

// SimplicialEGNNLayer fused edge-message + gated scatter kernel for gfx1250.
//
// Roofline: 43.3 GFLOP of fp32 GEMM vs ~90MB of L2-resident traffic ->
// compute bound -> per-edge GEMV through V_WMMA_F32_16X16X4_F32, K=264=66x4.
// Staging of gathered edge state uses GLOBAL_LOAD_ASYNC_TO_LDS_B128
// (ASYNCcnt-tracked DMA, no VGPR round-trip) double-buffered under the WMMA
// chain; completion via s_wait_asynccnt 0 + workgroup barrier.

#define H      128
#define NI     8
#define DIN    (2*H + NI)        // 264
#define TILE_M 16
#define STRIDE 268               // 264 + 4 pad: 16 rows -> 16 distinct banks, 16B aligned
#define NWAVES 8
#define BLOCK  (NWAVES*32)

typedef __attribute__((ext_vector_type(2))) float v2f;
typedef __attribute__((ext_vector_type(8))) float v8f;

// async-to-LDS builtin signatures (probe-confirmed via round-2 diagnostics):
//   b128: (int __vector(4) AS1*, int __vector(4) AS3*, imm int, imm int)
//   b32 : (int AS1*,             int AS3*,             imm int, imm int)
typedef int vs4i __attribute__((vector_size(16)));
typedef __attribute__((address_space(1))) vs4i* gv4_t;
typedef __attribute__((address_space(3))) vs4i* lv4_t;
typedef __attribute__((address_space(1))) int*  gi_t;
typedef __attribute__((address_space(3))) int*  li_t;

__device__ __forceinline__ void async_copy_b128(const void* g, void* l) {
#if __has_builtin(__builtin_amdgcn_global_load_async_to_lds_b128)
    __builtin_amdgcn_global_load_async_to_lds_b128(
        (gv4_t)(vs4i*)g, (lv4_t)(vs4i*)l, 0, 0);
#else
    unsigned lds = (unsigned)(uintptr_t)(li_t)(int*)l;
    asm volatile("global_load_async_to_lds_b128 %0, %1, off"
                 :: "v"(lds), "v"(g) : "memory");
#endif
}

__device__ __forceinline__ void async_copy_b32(const void* g, void* l) {
#if __has_builtin(__builtin_amdgcn_global_load_async_to_lds_b32)
    __builtin_amdgcn_global_load_async_to_lds_b32(
        (gi_t)(int*)g, (li_t)(int*)l, 0, 0);
#else
    unsigned lds = (unsigned)(uintptr_t)(li_t)(int*)l;
    asm volatile("global_load_async_to_lds_b32 %0, %1, off"
                 :: "v"(lds), "v"(g) : "memory");
#endif
}

__device__ __forceinline__ void wait_async0() {
#if __has_builtin(__builtin_amdgcn_s_wait_asynccnt)
    __builtin_amdgcn_s_wait_asynccnt(0);
#else
    asm volatile("s_wait_asynccnt 0" ::: "memory");
#endif
}

__global__ void zero_out_kernel(float* __restrict__ out, int n) {
    for (int i = blockIdx.x * blockDim.x + threadIdx.x; i < n;
         i += gridDim.x * blockDim.x)
        out[i] = 0.0f;
}

__global__ __launch_bounds__(BLOCK)
void egnn_edge_kernel(const float* __restrict__ x_send,
                      const float* __restrict__ x_rec,
                      const float* __restrict__ edge_attr,
                      const float* __restrict__ W1,
                      const float* __restrict__ b1,
                      const float* __restrict__ Wi,
                      const float* __restrict__ biP,
                      const int*   __restrict__ index_send,
                      const int*   __restrict__ index_rec,
                      float*       __restrict__ out,
                      int nTiles) {
    __shared__ float sWt[H * STRIDE];            // W1 transposed: [col][k], 134.2 KB
    __shared__ float sState[2][TILE_M * STRIDE]; // double-buffered edge state
    __shared__ float sRed[NWAVES][TILE_M];       // per-wave partial dots
    __shared__ float sWgt[TILE_M];               // per-edge sigmoid gate
    __shared__ int   sRec[2][TILE_M];            // per-edge receiver index

    const int tid  = threadIdx.x;
    const int lane = tid & 31;
    const int wave = tid >> 5;
    const int half = lane >> 4;          // 0: rows 0-7, 1: rows 8-15
    const int colL = lane & 15;
    const int col  = wave * 16 + colL;   // this wave's output column

    // ---- one-time: cache W1 transposed in LDS ----
    for (int idx = tid; idx < DIN * H; idx += BLOCK) {
        int k = idx >> 7;        // row of W1 (k in [0,264))
        int c = idx & (H - 1);   // col of W1
        sWt[c * STRIDE + k] = W1[idx];
    }

    const float bias = b1[col];
    const float wi   = Wi[col];
    const float biS  = biP[0];

    const int aOff = colL * STRIDE + (half << 1);
    const int bOff = col  * STRIDE + (half << 1);

    // ---- async staging of one tile's edge state into sState[nb] ----
    auto stage_async = [&](int tile, int nb) {
        const int base = tile * TILE_M;
        const int ew   = base + wave * 2;
        const int is0 = index_send[ew], is1 = index_send[ew + 1];
        const int ir0 = index_rec[ew],  ir1 = index_rec[ew + 1];
        float* row0 = &sState[nb][(wave * 2) * STRIDE];
        float* row1 = row0 + STRIDE;
        const int o = lane * 4;
        async_copy_b128(x_send + (size_t)is0 * H + o, row0 + o);
        async_copy_b128(x_send + (size_t)is1 * H + o, row1 + o);
        async_copy_b128(x_rec  + (size_t)ir0 * H + o, row0 + H + o);
        async_copy_b128(x_rec  + (size_t)ir1 * H + o, row1 + H + o);
        if (lane < 4) {
            int e = ew + (lane >> 1);
            float* rowE = (lane < 2) ? row0 : row1;
            async_copy_b128(edge_attr + (size_t)e * NI + (lane & 1) * 4,
                            rowE + 2 * H + (lane & 1) * 4);
        }
        if (wave == 0 && lane < TILE_M)
            async_copy_b32(index_rec + base + lane, &sRec[nb][lane]);
    };

    // ---- prologue: stage tile0 ----
    int tile = blockIdx.x;
    if (tile < nTiles) stage_async(tile, 0);
    wait_async0();
    __syncthreads();

    int cur = 0;
    for (; tile < nTiles; tile += gridDim.x) {
        const int nxt = tile + gridDim.x;

        // issue next tile's DMA; it lands in LDS while the WMMA chain runs
        if (nxt < nTiles) stage_async(nxt, cur ^ 1);

        // ---- GEMM: C(16x16) += A(16x264) * W1slab(264x16), 66 WMMA steps ----
        v8f c = {0.f, 0.f, 0.f, 0.f, 0.f, 0.f, 0.f, 0.f};
        #pragma unroll
        for (int k0 = 0; k0 < DIN; k0 += 4) {
            v2f a = *(const v2f*)(&sState[cur][aOff + k0]);
            v2f b = *(const v2f*)(&sWt[bOff + k0]);
            c = __builtin_amdgcn_wmma_f32_16x16x4_f32(
                    false, a, false, b, (short)0, c, false, false);
        }

        // ---- bias + SiLU, partial dot with Wi for the gate ----
        float part[8];
        #pragma unroll
        for (int r = 0; r < 8; ++r) {
            float x = c[r] + bias;
            float m = x / (1.0f + __expf(-x));   // silu
            c[r] = m;
            part[r] = m * wi;
        }
        // reduce across the 16 lanes of each half (cols of this wave's slab)
        #pragma unroll
        for (int off = 1; off < 16; off <<= 1) {
            #pragma unroll
            for (int r = 0; r < 8; ++r)
                part[r] += __shfl_xor(part[r], off, 32);
        }
        if (colL == 0) {
            #pragma unroll
            for (int r = 0; r < 8; ++r)
                sRed[wave][half * 8 + r] = part[r];
        }
        __syncthreads();

        if (tid < TILE_M) {
            float s = biS;
            #pragma unroll
            for (int w = 0; w < NWAVES; ++w) s += sRed[w][tid];
            sWgt[tid] = 1.0f / (1.0f + __expf(-s));   // sigmoid gate
        }
        __syncthreads();

        // ---- gated scatter-add into receivers ----
        #pragma unroll
        for (int r = 0; r < 8; ++r) {
            int row  = half * 8 + r;
            float w  = sWgt[row];
            int node = sRec[cur][row];
            atomicAdd(out + (size_t)node * H + col, c[r] * w);
        }

        // async staging for the next tile must have landed before publishing
        wait_async0();
        __syncthreads();
        cur ^= 1;
    }
}

extern "C" void kernel_launch(void* const* d_in, const int* in_sizes, int n_in,
                              void* d_out, int out_size, void* d_ws, size_t ws_size,
                              hipStream_t stream) {
    const float* x_send     = (const float*)d_in[0];
    const float* x_rec      = (const float*)d_in[1];
    const float* edge_attr  = (const float*)d_in[2];
    const float* W1         = (const float*)d_in[3];
    const float* b1         = (const float*)d_in[4];
    const float* Wi         = (const float*)d_in[5];
    const float* bi         = (const float*)d_in[6];
    const int*   index_send = (const int*)d_in[7];
    const int*   index_rec  = (const int*)d_in[8];
    float* out = (float*)d_out;

    const int nEdges = in_sizes[7];           // 640000
    const int nTiles = (nEdges + TILE_M - 1) / TILE_M;

    zero_out_kernel<<<512, 256, 0, stream>>>(out, out_size);
    egnn_edge_kernel<<<512, BLOCK, 0, stream>>>(
        x_send, x_rec, edge_attr, W1, b1, Wi, bi,
        index_send, index_rec, out, nTiles);
}